// InteractionBlock_39084202393769
// MI455X (gfx1250) — compile-verified
//
#include <hip/hip_runtime.h>
#include <math.h>
#include <stdint.h>

#define N_NODES 50000
#define N_EDGES 1600000
#define HID 128
#define NRBF 50
#define RPAD 52                 // 50 padded to multiple of 4 for K-steps
#define PI_OVER_CUTOFF 0.31415926535897931f
#define CUTOFF 10.0f

#if defined(__has_builtin)
#if __has_builtin(__builtin_amdgcn_global_load_async_to_lds_b64) && \
    __has_builtin(__builtin_amdgcn_global_load_async_to_lds_b32)
#define HAVE_ASYNC_LDS 1
#endif
#if __has_builtin(__builtin_amdgcn_s_wait_asynccnt)
#define HAVE_WAIT_ASYNC 1
#endif
#endif
#ifndef HAVE_ASYNC_LDS
#define HAVE_ASYNC_LDS 0
#endif
#ifndef HAVE_WAIT_ASYNC
#define HAVE_WAIT_ASYNC 0
#endif

typedef __attribute__((ext_vector_type(2))) float v2f;
typedef __attribute__((ext_vector_type(8))) float v8f;
typedef int v2i __attribute__((vector_size(8)));   // matches builtin param type

template <int N>
__device__ __forceinline__ void wait_asynccnt_le() {
#if HAVE_WAIT_ASYNC
  __builtin_amdgcn_s_wait_asynccnt(N);
#else
  asm volatile("s_wait_asynccnt %0" ::"i"(N) : "memory");
#endif
}

// One 16x16 f32 output tile accumulated over K via V_WMMA_F32_16X16X4_F32.
// A: 16xK row-major (lda), B: Kx16 row-major (ldb), both expected in LDS.
// ISA VGPR layouts (CDNA5 ISA §7.12.2):
//   A 16x4 :  lane[3:0]=M, lane[4] selects K-pair {0,1}/{2,3}, 2 VGPRs
//   B 4x16 :  lane[3:0]=N, lane[4] selects K-pair, 2 VGPRs
//   C/D    :  VGPR v <-> (M = v + 8*lane[4], N = lane[3:0])
template <int K>
__device__ __forceinline__ v8f wmma_ktile(const float* __restrict__ A, int lda,
                                          const float* __restrict__ B, int ldb,
                                          v8f acc) {
  const int lane = threadIdx.x & 31;
  const int mn = lane & 15;
  const int ks = (lane >> 4) << 1;
#pragma unroll
  for (int k = 0; k < K; k += 4) {
    v2f a, b;
    a.x = A[mn * lda + k + ks];
    a.y = A[mn * lda + k + ks + 1];
    b.x = B[(k + ks) * ldb + mn];
    b.y = B[(k + ks + 1) * ldb + mn];
    acc = __builtin_amdgcn_wmma_f32_16x16x4_f32(false, a, false, b,
                                                (short)0, acc, false, false);
  }
  return acc;
}

// ---------------- Kernel 1: h = x @ lin1_w ; agg = 0 -----------------------
extern "C" __global__ void __launch_bounds__(256)
k_node_pre(const float* __restrict__ x, const float* __restrict__ w,
           float* __restrict__ h, float* __restrict__ agg) {
  extern __shared__ float smem[];
  float* Ws = smem;                 // HID*HID
  float* Xs = smem + HID * HID;     // 16*HID
  const int tid = threadIdx.x;
  const int row0 = blockIdx.x * 16; // N_NODES is an exact multiple of 16

  for (int i = tid; i < HID * HID / 4; i += 256)
    ((float4*)Ws)[i] = ((const float4*)w)[i];
  for (int i = tid; i < 16 * HID / 4; i += 256)
    ((float4*)Xs)[i] = ((const float4*)(x + (size_t)row0 * HID))[i];
  for (int i = tid; i < 16 * HID / 4; i += 256)
    ((float4*)(agg + (size_t)row0 * HID))[i] = make_float4(0.f, 0.f, 0.f, 0.f);
  __syncthreads();

  const int wave = tid >> 5;
  const int n0 = wave * 16;
  v8f acc = {};
  acc = wmma_ktile<HID>(Xs, HID, Ws + n0, HID, acc);

  const int lane = tid & 31;
  const int n = lane & 15;
  const int mh = (lane >> 4) * 8;
#pragma unroll
  for (int v = 0; v < 8; ++v)
    h[(size_t)(row0 + mh + v) * HID + n0 + n] = acc[v];
}

// ---- Kernel 2: fused filter-net + cutoff + gather + modulate + scatter ----
#if HAVE_ASYNC_LDS
// Per-wave async batch is EXACTLY 5 instructions:
//   2x B64 (one EA row each, lanes 0..24) + 3x B32 meta (lanes 0..15).
// Uniform count => s_wait_asynccnt 5 after issuing the next batch drains
// precisely the previous batch (async loads complete in order).
__device__ __forceinline__ void issue_tile(
    const float* __restrict__ edge_attr, const float* __restrict__ edge_weight,
    const int* __restrict__ edge_index, int e0,
    float* EA, float* EW, int* SRC, int* DST, int lane, int wave) {
#pragma unroll
  for (int rr = 0; rr < 2; ++rr) {
    const int r = 2 * wave + rr;
    if (lane < 25) {
      const float* g = edge_attr + (size_t)(e0 + r) * NRBF + 2 * lane;
      float* l = EA + r * RPAD + 2 * lane;
      __builtin_amdgcn_global_load_async_to_lds_b64((v2i*)g, (v2i*)l, 0, 0);
    }
  }
  if (lane < 16) {
    __builtin_amdgcn_global_load_async_to_lds_b32(
        (int*)(edge_weight + e0 + lane), (int*)(EW + lane), 0, 0);
    __builtin_amdgcn_global_load_async_to_lds_b32(
        (int*)(edge_index + e0 + lane), SRC + lane, 0, 0);
    __builtin_amdgcn_global_load_async_to_lds_b32(
        (int*)(edge_index + N_EDGES + e0 + lane), DST + lane, 0, 0);
  }
}
#endif

extern "C" __global__ void __launch_bounds__(256)
k_edge(const float* __restrict__ edge_attr, const float* __restrict__ edge_weight,
       const int* __restrict__ edge_index,
       const float* __restrict__ fw1_w, const float* __restrict__ fw1_b,
       const float* __restrict__ fw2_w, const float* __restrict__ fw2_b,
       const float* __restrict__ h, float* __restrict__ agg) {
  extern __shared__ float smem[];
  float* F1  = smem;                  // RPAD*HID (rows >= 50 zeroed)
  float* F2  = F1 + RPAD * HID;       // HID*HID
  float* B1  = F2 + HID * HID;        // HID
  float* B2  = B1 + HID;              // HID
  float* T   = B2 + HID;              // 16*HID  (tanh of stage-1)
  float* EA0 = T + 16 * HID;          // 16*RPAD double-buffered tile
  float* EA1 = EA0 + 16 * RPAD;       // 16*RPAD
  float* EW0 = EA1 + 16 * RPAD;       // 16
  float* EW1 = EW0 + 16;              // 16
  float* CV  = EW1 + 16;              // 16
  int* SRC0  = (int*)(CV + 16);       // 16
  int* DST0  = SRC0 + 16;             // 16
  int* SRC1  = DST0 + 16;             // 16
  int* DST1  = SRC1 + 16;             // 16

  const int tid = threadIdx.x;
  const int wave = tid >> 5;
  const int lane = tid & 31;

  // Resident filter weights: loaded once per persistent block.
  for (int i = tid; i < RPAD * HID; i += 256) {
    const int k = i / HID;
    F1[i] = (k < NRBF) ? fw1_w[i] : 0.0f;
  }
  for (int i = tid; i < HID * HID / 4; i += 256)
    ((float4*)F2)[i] = ((const float4*)fw2_w)[i];
  if (tid < HID) { B1[tid] = fw1_b[tid]; B2[tid] = fw2_b[tid]; }
  // Zero-pad columns 50..51 of both EA buffers once (DMA never touches them).
  if (tid < 32) {
    float* e = (tid & 16) ? EA1 : EA0;
    const int r = tid & 15;
    e[r * RPAD + NRBF] = 0.0f;
    e[r * RPAD + NRBF + 1] = 0.0f;
  }

  const int n = lane & 15;
  const int mh = (lane >> 4) * 8;
  const int n0 = wave * 16;
  const int ntiles = N_EDGES / 16;   // exact
  const int G = gridDim.x;

#if HAVE_ASYNC_LDS
  if (blockIdx.x < ntiles)
    issue_tile(edge_attr, edge_weight, edge_index, blockIdx.x * 16,
               EA0, EW0, SRC0, DST0, lane, wave);
#endif
  __syncthreads();

  int parity = 0;
  for (int t = blockIdx.x; t < ntiles; t += G) {
    float* EAc = parity ? EA1 : EA0;
    float* EWc = parity ? EW1 : EW0;
    int* SRCc  = parity ? SRC1 : SRC0;
    int* DSTc  = parity ? DST1 : DST0;

#if HAVE_ASYNC_LDS
    const bool more = (t + G) < ntiles;
    if (more)
      issue_tile(edge_attr, edge_weight, edge_index, (t + G) * 16,
                 parity ? EA0 : EA1, parity ? EW0 : EW1,
                 parity ? SRC0 : SRC1, parity ? DST0 : DST1, lane, wave);
    if (more) wait_asynccnt_le<5>();   // drain previous batch only
    else      wait_asynccnt_le<0>();   // last tile: drain everything
#else
    const int e0 = t * 16;
    for (int i = tid; i < 16 * RPAD; i += 256) {
      const int r = i / RPAD, c = i % RPAD;
      if (c < NRBF) EAc[i] = edge_attr[(size_t)(e0 + r) * NRBF + c];
    }
    if (tid < 16) {
      EWc[tid] = edge_weight[e0 + tid];
      SRCc[tid] = edge_index[e0 + tid];
      DSTc[tid] = edge_index[N_EDGES + e0 + tid];
    }
#endif
    __syncthreads();

    // Cutoff envelope + warm the h[src] rows while stage 1 runs.
    if (tid < 16) {
      const float d = EWc[tid];
      CV[tid] = (d < CUTOFF) ? 0.5f * (cosf(d * PI_OVER_CUTOFF) + 1.0f) : 0.0f;
    }
    if (tid < 64)
      __builtin_prefetch(h + (size_t)SRCc[tid >> 2] * HID + (tid & 3) * 32, 0, 0);

    // Stage 1: T = tanh(EA @ F1 + b1), each wave owns one 16-col tile.
    v8f a1 = {};
    {
      const float bb = B1[n0 + n];
#pragma unroll
      for (int v = 0; v < 8; ++v) a1[v] = bb;
    }
    a1 = wmma_ktile<RPAD>(EAc, RPAD, F1 + n0, HID, a1);
#pragma unroll
    for (int v = 0; v < 8; ++v)
      T[(mh + v) * HID + n0 + n] = tanhf(a1[v]);
    __syncthreads();

    // Stage 2: W = T @ F2 + b2; scale by cutoff, gather h[src], scatter.
    v8f a2 = {};
    {
      const float bb = B2[n0 + n];
#pragma unroll
      for (int v = 0; v < 8; ++v) a2[v] = bb;
    }
    a2 = wmma_ktile<HID>(T, HID, F2 + n0, HID, a2);
#pragma unroll
    for (int v = 0; v < 8; ++v) {
      const int m = mh + v;
      const float msg = a2[v] * CV[m] * h[(size_t)SRCc[m] * HID + n0 + n];
      unsafeAtomicAdd(&agg[(size_t)DSTc[m] * HID + n0 + n], msg);
    }
    __syncthreads();   // protect T / staged buffers before next iteration
    parity ^= 1;
  }
}

// -------- Kernel 3: out = tanh(agg @ lin2 + b2) @ lin + b ------------------
extern "C" __global__ void __launch_bounds__(256)
k_epilogue(const float* __restrict__ agg,
           const float* __restrict__ lin2_w, const float* __restrict__ lin2_b,
           const float* __restrict__ lin_w, const float* __restrict__ lin_b,
           float* __restrict__ out) {
  extern __shared__ float smem[];
  float* W2  = smem;                  // HID*HID
  float* W3  = W2 + HID * HID;        // HID*HID
  float* B2v = W3 + HID * HID;        // HID
  float* B3v = B2v + HID;             // HID
  float* AG  = B3v + HID;             // 16*HID
  float* T   = AG + 16 * HID;         // 16*HID

  const int tid = threadIdx.x;
  const int row0 = blockIdx.x * 16;

  for (int i = tid; i < HID * HID / 4; i += 256) {
    ((float4*)W2)[i] = ((const float4*)lin2_w)[i];
    ((float4*)W3)[i] = ((const float4*)lin_w)[i];
  }
  if (tid < HID) { B2v[tid] = lin2_b[tid]; B3v[tid] = lin_b[tid]; }
  for (int i = tid; i < 16 * HID / 4; i += 256)
    ((float4*)AG)[i] = ((const float4*)(agg + (size_t)row0 * HID))[i];
  __syncthreads();

  const int wave = tid >> 5;
  const int lane = tid & 31;
  const int n = lane & 15;
  const int mh = (lane >> 4) * 8;
  const int n0 = wave * 16;

  // y = agg @ lin2 + b2 ; T = tanh(y)
  v8f acc = {};
  {
    const float bb = B2v[n0 + n];
#pragma unroll
    for (int v = 0; v < 8; ++v) acc[v] = bb;
  }
  acc = wmma_ktile<HID>(AG, HID, W2 + n0, HID, acc);
#pragma unroll
  for (int v = 0; v < 8; ++v)
    T[(mh + v) * HID + n0 + n] = tanhf(acc[v]);
  __syncthreads();

  // out = T @ lin_w + b
  v8f acc2 = {};
  {
    const float bb = B3v[n0 + n];
#pragma unroll
    for (int v = 0; v < 8; ++v) acc2[v] = bb;
  }
  acc2 = wmma_ktile<HID>(T, HID, W3 + n0, HID, acc2);
#pragma unroll
  for (int v = 0; v < 8; ++v)
    out[(size_t)(row0 + mh + v) * HID + n0 + n] = acc2[v];
}

extern "C" void kernel_launch(void* const* d_in, const int* in_sizes, int n_in,
                              void* d_out, int out_size, void* d_ws, size_t ws_size,
                              hipStream_t stream) {
  const float* x      = (const float*)d_in[0];
  const int*   ei     = (const int*)d_in[1];   // [2, E] int32 (JAX x64-off)
  const float* ew     = (const float*)d_in[2];
  const float* ea     = (const float*)d_in[3];
  const float* lin1_w = (const float*)d_in[4];
  const float* fw1_w  = (const float*)d_in[5];
  const float* fw1_b  = (const float*)d_in[6];
  const float* fw2_w  = (const float*)d_in[7];
  const float* fw2_b  = (const float*)d_in[8];
  const float* lin2_w = (const float*)d_in[9];
  const float* lin2_b = (const float*)d_in[10];
  const float* lin_w  = (const float*)d_in[11];
  const float* lin_b  = (const float*)d_in[12];
  float* out = (float*)d_out;

  float* h   = (float*)d_ws;                        // [N, HID]
  float* agg = h + (size_t)N_NODES * HID;           // [N, HID]

  const size_t smemA = (size_t)(HID * HID + 16 * HID) * sizeof(float);
  const size_t smemB = (size_t)(RPAD * HID + HID * HID + 2 * HID + 16 * HID +
                                2 * 16 * RPAD + 3 * 16) * sizeof(float) +
                       64 * sizeof(int);
  const size_t smemC = (size_t)(2 * HID * HID + 2 * HID + 2 * 16 * HID) * sizeof(float);

  (void)hipFuncSetAttribute((const void*)k_node_pre,
                            hipFuncAttributeMaxDynamicSharedMemorySize, (int)smemA);
  (void)hipFuncSetAttribute((const void*)k_edge,
                            hipFuncAttributeMaxDynamicSharedMemorySize, (int)smemB);
  (void)hipFuncSetAttribute((const void*)k_epilogue,
                            hipFuncAttributeMaxDynamicSharedMemorySize, (int)smemC);

  k_node_pre<<<N_NODES / 16, 256, smemA, stream>>>(x, lin1_w, h, agg);
  k_edge<<<1024, 256, smemB, stream>>>(ea, ew, ei, fw1_w, fw1_b,
                                       fw2_w, fw2_b, h, agg);
  k_epilogue<<<N_NODES / 16, 256, smemC, stream>>>(agg, lin2_w, lin2_b,
                                                   lin_w, lin_b, out);
}